// lightGCN_3032246911311
// MI455X (gfx1250) — compile-verified
//
#include <hip/hip_runtime.h>
#include <hip/hip_bf16.h>
#include <cstdint>

// ---------------------------------------------------------------------------
// LightGCN propagation for MI455X (gfx1250).
//   ego0 = concat(user_emb, item_emb)           (150000 x 64 f32)
//   3x:  ego = segment_sum(edge_val * ego[col], row);  acc += ego
//   out  = gather(acc * 0.25) for users / pos / neg    (3 x 4096 x 64)
// Memory-bound scatter/gather; embedding tables are L2-resident (38.4MB/buf).
// CDNA5 paths: 4-deep async global->LDS ring (ASYNCcnt) for the random row
// gather, ds_load_b128 consume, non-temporal edge streams, f32 global atomics.
// ---------------------------------------------------------------------------

#define LG_N_USERS  100000
#define LG_N_ITEMS  50000
#define LG_N_NODES  (LG_N_USERS + LG_N_ITEMS)
#define LG_DIM      64
#define LG_LAYERS   3

#if defined(__has_builtin)
# if __has_builtin(__builtin_amdgcn_global_load_async_to_lds_b128) && \
     __has_builtin(__builtin_amdgcn_s_wait_asynccnt)
#  define LG_ASYNC 1
# endif
#endif
#ifndef LG_ASYNC
# define LG_ASYNC 0
#endif

#if LG_ASYNC
// Exact pointee type from the builtin's signature:
//   'int __attribute__((vector_size(16))) __device__ *'
typedef int lg_v4i __attribute__((vector_size(16)));
typedef __attribute__((address_space(1))) lg_v4i* lg_gptr_t;  // global
typedef __attribute__((address_space(3))) lg_v4i* lg_sptr_t;  // LDS
#endif

// ---- helpers --------------------------------------------------------------

__device__ __forceinline__ void lg_issue_load16(const float* g, float* lds) {
#if LG_ASYNC
  // global (as1) -> LDS (as3) 16-byte async copy, tracked by ASYNCcnt.
  __builtin_amdgcn_global_load_async_to_lds_b128(
      (lg_gptr_t)(uintptr_t)g,
      (lg_sptr_t)(uint32_t)(uintptr_t)lds,
      0, 0);
#else
  *(float4*)lds = *(const float4*)g;
#endif
}

// Wait until at most `left` async ops remain outstanding (block-uniform).
__device__ __forceinline__ void lg_wait_async_upto(long long left) {
#if LG_ASYNC
  if (left <= 0)      __builtin_amdgcn_s_wait_asynccnt(0);
  else if (left == 1) __builtin_amdgcn_s_wait_asynccnt(1);
  else if (left == 2) __builtin_amdgcn_s_wait_asynccnt(2);
  else                __builtin_amdgcn_s_wait_asynccnt(3);
#else
  (void)left;
#endif
}

__device__ __forceinline__ void lg_wait_ds0() {
#if LG_ASYNC
# if defined(__has_builtin)
#  if __has_builtin(__builtin_amdgcn_s_wait_dscnt)
  __builtin_amdgcn_s_wait_dscnt(0);
#  endif
# endif
#endif
}

__device__ __forceinline__ void lg_atom_add(float* p, float v) {
  // lowers to global_atomic_add_f32 (native, no CAS loop) on CDNA
  unsafeAtomicAdd(p, v);
}

// ---- kernels --------------------------------------------------------------

// ego = src, acc = src, nxt = 0
__global__ void lg_init(const float4* __restrict__ user4,
                        const float4* __restrict__ item4,
                        float4* __restrict__ ego,
                        float4* __restrict__ acc,
                        float4* __restrict__ nxt,
                        int user4n, int total4)
{
  int i = blockIdx.x * blockDim.x + threadIdx.x;
  if (i >= total4) return;
  float4 e = (i < user4n) ? user4[i] : item4[i - user4n];
  ego[i] = e;
  acc[i] = e;
  nxt[i] = make_float4(0.f, 0.f, 0.f, 0.f);
}

// dst[row] += val * src[col]  (one edge per 16 lanes; float4 per lane)
// Grid-stride over 16-edge chunks through a 4-deep async LDS ring:
// per wave, up to 4 random row-gathers are in flight (ASYNCcnt), hiding
// L2/HBM latency while the consume side issues fire-and-forget atomics.
// Each wave owns 2 edges per chunk and only touches its own LDS slices,
// so no workgroup barrier is needed — only per-wave counter waits.
__launch_bounds__(256)
__global__ void lg_scatter(const int*   __restrict__ erow,
                           const int*   __restrict__ ecol,
                           const float* __restrict__ eval,
                           const float* __restrict__ src,
                           float*       __restrict__ dst,
                           int nnz)
{
  __shared__ __align__(16) float buf[4][16 * LG_DIM];  // 16 KB, 4-slot ring
  const int tid = threadIdx.x;
  const int le  = tid >> 4;        // local edge 0..15
  const int sub = tid & 15;        // float4 slot within the 64-dim row
  float* slotBase = &buf[0][le * LG_DIM + sub * 4];
  const int S = 16 * LG_DIM;       // floats between ring slots

  const long long nchunks = ((long long)nnz + 15) >> 4;
  const long long G = gridDim.x;
  const long long c0 = blockIdx.x;
  if (c0 >= nchunks) return;

  // Per-slot in-flight metadata, statically rotated (no dynamic indexing).
  int   r0 = 0, r1 = 0, r2 = 0, r3 = 0;
  float v0 = 0.f, v1 = 0.f, v2 = 0.f, v3 = 0.f;
  bool  k0 = false, k1 = false, k2 = false, k3 = false;

  // Fetch edge metadata (non-temporal: streamed once, keep L2 for tables)
  // and issue the async row gather for chunk `ck` into ring slot `s`.
  auto issue = [&](long long ck, int s, int& rO, float& vO, bool& kO) {
    long long e  = ck * 16 + le;
    kO           = (e < (long long)nnz);
    long long ec = kO ? e : (long long)nnz - 1;   // clamp: EXEC stays full
    int col = __builtin_nontemporal_load(ecol + ec);
    rO      = __builtin_nontemporal_load(erow + ec);
    vO      = __builtin_nontemporal_load(eval + ec);
    lg_issue_load16(src + (long long)col * LG_DIM + sub * 4, slotBase + s * S);
  };

  auto consume = [&](int s, int rr, float vv, bool kk) {
    float4 m = *(const float4*)(slotBase + s * S);
    if (kk) {
      float* d = dst + (long long)rr * LG_DIM + sub * 4;
      lg_atom_add(d + 0, vv * m.x);
      lg_atom_add(d + 1, vv * m.y);
      lg_atom_add(d + 2, vv * m.z);
      lg_atom_add(d + 3, vv * m.w);
    }
  };

  // Prologue: fill the pipeline with up to 4 chunks.
  issue(c0, 0, r0, v0, k0);
  if (c0 + 1 * G < nchunks) issue(c0 + 1 * G, 1, r1, v1, k1);
  if (c0 + 2 * G < nchunks) issue(c0 + 2 * G, 2, r2, v2, k2);
  if (c0 + 3 * G < nchunks) issue(c0 + 3 * G, 3, r3, v3, k3);

  long long ck = c0;
  while (true) {
    // ---- stage: slot 0 ----
    {
      lg_wait_async_upto((nchunks - 1 - ck) / G);   // chunk ck is in LDS
      consume(0, r0, v0, k0);
      long long cf = ck + 4 * G;
      if (cf < nchunks) { lg_wait_ds0(); issue(cf, 0, r0, v0, k0); }
      ck += G; if (ck >= nchunks) break;
    }
    // ---- stage: slot 1 ----
    {
      lg_wait_async_upto((nchunks - 1 - ck) / G);
      consume(1, r1, v1, k1);
      long long cf = ck + 4 * G;
      if (cf < nchunks) { lg_wait_ds0(); issue(cf, 1, r1, v1, k1); }
      ck += G; if (ck >= nchunks) break;
    }
    // ---- stage: slot 2 ----
    {
      lg_wait_async_upto((nchunks - 1 - ck) / G);
      consume(2, r2, v2, k2);
      long long cf = ck + 4 * G;
      if (cf < nchunks) { lg_wait_ds0(); issue(cf, 2, r2, v2, k2); }
      ck += G; if (ck >= nchunks) break;
    }
    // ---- stage: slot 3 ----
    {
      lg_wait_async_upto((nchunks - 1 - ck) / G);
      consume(3, r3, v3, k3);
      long long cf = ck + 4 * G;
      if (cf < nchunks) { lg_wait_ds0(); issue(cf, 3, r3, v3, k3); }
      ck += G; if (ck >= nchunks) break;
    }
  }
}

// acc += nxt; zero out ego_old (it becomes the next scatter target)
__global__ void lg_accum_zero(float4* __restrict__ acc,
                              const float4* __restrict__ nxt,
                              float4* __restrict__ ego_old,
                              int total4)
{
  int i = blockIdx.x * blockDim.x + threadIdx.x;
  if (i >= total4) return;
  float4 a = acc[i];
  float4 n = nxt[i];
  a.x += n.x; a.y += n.y; a.z += n.z; a.w += n.w;
  acc[i] = a;
  ego_old[i] = make_float4(0.f, 0.f, 0.f, 0.f);
}

// out = 0.25 * acc[gathered rows] for users / pos_items / neg_items
__global__ void lg_gather(const float* __restrict__ acc,
                          const int* __restrict__ users,
                          const int* __restrict__ pos,
                          const int* __restrict__ neg,
                          float4* __restrict__ out,
                          int batch, int n_users)
{
  int t = blockIdx.x * blockDim.x + threadIdx.x;
  int per = batch * (LG_DIM / 4);
  int total = 3 * per;
  if (t >= total) return;
  int sec = t / per;
  int r   = t - sec * per;
  int b   = r >> 4;
  int sub = r & 15;
  int node;
  if      (sec == 0) node = users[b];
  else if (sec == 1) node = n_users + pos[b];
  else               node = n_users + neg[b];
  float4 m = ((const float4*)acc)[(long long)node * (LG_DIM / 4) + sub];
  out[t] = make_float4(m.x * 0.25f, m.y * 0.25f, m.z * 0.25f, m.w * 0.25f);
}

// ---- host launch ----------------------------------------------------------

extern "C" void kernel_launch(void* const* d_in, const int* in_sizes, int n_in,
                              void* d_out, int out_size, void* d_ws, size_t ws_size,
                              hipStream_t stream)
{
  (void)n_in; (void)out_size; (void)ws_size;
  const int*   erow = (const int*)  d_in[0];
  const int*   ecol = (const int*)  d_in[1];
  const float* eval = (const float*)d_in[2];
  const float* uemb = (const float*)d_in[3];
  const float* iemb = (const float*)d_in[4];
  const int*   users = (const int*) d_in[5];
  const int*   positems = (const int*)d_in[6];
  const int*   negitems = (const int*)d_in[7];

  const int nnz   = in_sizes[0];
  const int batch = in_sizes[5];

  const size_t nodeElems = (size_t)LG_N_NODES * LG_DIM;  // 9.6M floats
  float* bufA = (float*)d_ws;               // ego / next (ping-pong)
  float* bufB = bufA + nodeElems;
  float* accb = bufB + nodeElems;           // running accumulator

  const int total4 = (int)(nodeElems / 4);                    // 2.4M float4
  const int user4n = LG_N_USERS * LG_DIM / 4;

  dim3 blk(256);
  const int initBlocks = (total4 + 255) / 256;
  lg_init<<<initBlocks, blk, 0, stream>>>(
      (const float4*)uemb, (const float4*)iemb,
      (float4*)bufA, (float4*)accb, (float4*)bufB, user4n, total4);

  float* ego = bufA;
  float* nxt = bufB;
  const long long nchunks = ((long long)nnz + 15) >> 4;
  const int scatterBlocks = (int)((nchunks < 6144) ? nchunks : 6144);
  for (int l = 0; l < LG_LAYERS; ++l) {
    lg_scatter<<<scatterBlocks, blk, 0, stream>>>(erow, ecol, eval, ego, nxt, nnz);
    lg_accum_zero<<<initBlocks, blk, 0, stream>>>(
        (float4*)accb, (const float4*)nxt, (float4*)ego, total4);
    float* t = ego; ego = nxt; nxt = t;
  }

  const int gtotal  = 3 * batch * (LG_DIM / 4);
  const int gBlocks = (gtotal + 255) / 256;
  lg_gather<<<gBlocks, blk, 0, stream>>>(
      accb, users, positems, negitems, (float4*)d_out, batch, LG_N_USERS);
}